// GraphAttentionNetwork_39934605918424
// MI455X (gfx1250) — compile-verified
//
#include <hip/hip_runtime.h>
#include <hip/hip_bf16.h>

typedef float v2f __attribute__((ext_vector_type(2)));
typedef float v8f __attribute__((ext_vector_type(8)));

#define IN_CH   128
#define OUT_CH  128
#define HEADS   4
#define CH      32
#define NEG_SLOPE 0.2f

// ---------------------------------------------------------------------------
// monotonic float <-> uint key (order preserving) for atomicMax-based segmax
// ---------------------------------------------------------------------------
__device__ __forceinline__ unsigned fkey(float f) {
    unsigned u = __float_as_uint(f);
    return (u & 0x80000000u) ? ~u : (u | 0x80000000u);
}
__device__ __forceinline__ float fdec(unsigned k) {
    unsigned u = (k & 0x80000000u) ? (k ^ 0x80000000u) : ~k;
    return __uint_as_float(u);
}

// ---------------------------------------------------------------------------
// K0: out = bias (broadcast), amax keys = 0 (== -inf sentinel), denom = 0
// ---------------------------------------------------------------------------
__global__ void gat_init(float* __restrict__ out, const float* __restrict__ bias,
                         unsigned* __restrict__ amax, float* __restrict__ denom, int n) {
    int i = blockIdx.x * blockDim.x + threadIdx.x;
    if (i < n * OUT_CH) out[i] = bias[i & (OUT_CH - 1)];
    if (i < n * HEADS) { amax[i] = 0u; denom[i] = 0.0f; }
}

// ---------------------------------------------------------------------------
// K1: h = x @ W via V_WMMA_F32_16X16X4_F32.
// Block = 256 threads = 8 waves; W (128x128 f32, 64KB) staged in LDS.
// Each wave produces a 16x128 tile of h: 8 column-tile accumulators, 32 K-steps.
// A frag (16x4):  lanes 0-15 -> K = {4k,4k+1}, lanes 16-31 -> K = {4k+2,4k+3}
// B frag (4x16):  same K split across lane halves, N = lane&15
// D (16x16): VGPR j -> row j (lanes 0-15) / row 8+j (lanes 16-31), col = lane&15
// ---------------------------------------------------------------------------
__global__ __launch_bounds__(256) void gat_gemm(const float* __restrict__ x,
                                                const float* __restrict__ W,
                                                float* __restrict__ h, int n) {
    __shared__ float Wl[IN_CH * OUT_CH];
    for (int i = threadIdx.x; i < IN_CH * OUT_CH; i += 256) Wl[i] = W[i];
    __syncthreads();

    const int wave = threadIdx.x >> 5;
    const int lane = threadIdx.x & 31;
    const int m    = lane & 15;          // row-in-tile (A) / col-in-tile (B,D)
    const int koff = (lane >> 4) << 1;   // 0 or 2
    const int rowbase = blockIdx.x * 128 + wave * 16;

    int lrow = rowbase + m;
    if (lrow > n - 1) lrow = n - 1;      // clamp loads: EXEC must stay all-1s for WMMA
    const float* xr = x + (size_t)lrow * IN_CH + koff;

    v8f acc[8] = {};
    for (int k = 0; k < IN_CH; k += 4) {
        v2f a;
        a.x = xr[k];
        a.y = xr[k + 1];
        const float* wr = &Wl[(k + koff) * OUT_CH + m];
#pragma unroll
        for (int t = 0; t < 8; ++t) {
            v2f b;
            b.x = wr[t * 16];
            b.y = wr[OUT_CH + t * 16];
            acc[t] = __builtin_amdgcn_wmma_f32_16x16x4_f32(
                false, a, false, b, (short)0, acc[t], false, false);
        }
    }

    const int orow0 = rowbase + ((lane >> 4) << 3);
#pragma unroll
    for (int j = 0; j < 8; ++j) {
        int orow = orow0 + j;
        if (orow < n) {
            float* hp = h + (size_t)orow * OUT_CH + m;
#pragma unroll
            for (int t = 0; t < 8; ++t) hp[t * 16] = acc[t][j];
        }
    }
}

// ---------------------------------------------------------------------------
// K2: per-node attention logits  a_src[n,hd] = <h[n,hd,:], att_src[hd,:]>
// ---------------------------------------------------------------------------
__global__ void gat_logits(const float* __restrict__ h,
                           const float* __restrict__ att_src,
                           const float* __restrict__ att_dst,
                           float* __restrict__ a_src, float* __restrict__ a_dst, int n) {
    int i = blockIdx.x * blockDim.x + threadIdx.x;
    if (i >= n * HEADS) return;
    int hd = i & (HEADS - 1);
    const float* hp = h + (size_t)(i >> 2) * OUT_CH + hd * CH;
    const float* as = att_src + hd * CH;
    const float* ad = att_dst + hd * CH;
    float ss = 0.f, sd = 0.f;
#pragma unroll
    for (int c = 0; c < CH; ++c) {
        float v = hp[c];
        ss += v * as[c];
        sd += v * ad[c];
    }
    a_src[i] = ss;
    a_dst[i] = sd;
}

__device__ __forceinline__ void edge_sd(const int* __restrict__ ei, int idx, int e,
                                        int& s, int& d) {
    if (idx < e) { s = ei[idx]; d = ei[e + idx]; }
    else         { s = d = idx - e; }        // self loops appended
}

// ---------------------------------------------------------------------------
// K3: segment max over incoming edges (uint-key atomicMax)
// ---------------------------------------------------------------------------
__global__ void gat_pass1(const int* __restrict__ ei,
                          const float* __restrict__ a_src, const float* __restrict__ a_dst,
                          unsigned* __restrict__ amax, int e, int n) {
    int idx = blockIdx.x * blockDim.x + threadIdx.x;
    if (idx >= e + n) return;
    int s, d;
    edge_sd(ei, idx, e, s, d);
    const float4 as = ((const float4*)a_src)[s];
    const float4 ad = ((const float4*)a_dst)[d];
    float al[HEADS] = {as.x + ad.x, as.y + ad.y, as.z + ad.z, as.w + ad.w};
#pragma unroll
    for (int hd = 0; hd < HEADS; ++hd) {
        float a = al[hd];
        a = a > 0.f ? a : NEG_SLOPE * a;
        atomicMax(&amax[d * HEADS + hd], fkey(a));
    }
}

// ---------------------------------------------------------------------------
// K4: ex = exp(alpha - amax[dst]); denom[dst] += ex  (ex stored per edge)
// ---------------------------------------------------------------------------
__global__ void gat_pass2(const int* __restrict__ ei,
                          const float* __restrict__ a_src, const float* __restrict__ a_dst,
                          const unsigned* __restrict__ amax,
                          float* __restrict__ exbuf, float* __restrict__ denom,
                          int e, int n) {
    int idx = blockIdx.x * blockDim.x + threadIdx.x;
    if (idx >= e + n) return;
    int s, d;
    edge_sd(ei, idx, e, s, d);
    const float4 as = ((const float4*)a_src)[s];
    const float4 ad = ((const float4*)a_dst)[d];
    float al[HEADS] = {as.x + ad.x, as.y + ad.y, as.z + ad.z, as.w + ad.w};
    float4 exo;
    float* exp_ptr = (float*)&exo;
#pragma unroll
    for (int hd = 0; hd < HEADS; ++hd) {
        float a = al[hd];
        a = a > 0.f ? a : NEG_SLOPE * a;
        float mx = fdec(amax[d * HEADS + hd]);
        float ex = __expf(a - mx);
        exp_ptr[hd] = ex;
        atomicAdd(&denom[d * HEADS + hd], ex);
    }
    ((float4*)exbuf)[idx] = exo;
}

// ---------------------------------------------------------------------------
// K5: message pass — one wave per edge; lane l handles channels 4l..4l+3.
// out[dst] += coef[head] * h[src]   (float atomics)
// ---------------------------------------------------------------------------
__global__ __launch_bounds__(256) void gat_pass3(const int* __restrict__ ei,
                                                 const float* __restrict__ h,
                                                 const float* __restrict__ exbuf,
                                                 const float* __restrict__ denom,
                                                 float* __restrict__ out, int e, int n) {
    int gid  = blockIdx.x * blockDim.x + threadIdx.x;
    int eidx = gid >> 5;
    int lane = gid & 31;
    if (eidx >= e + n) return;
    int s, d;
    edge_sd(ei, eidx, e, s, d);
    int hd = lane >> 3;                                // 8 lanes per head
    float coef = exbuf[(size_t)eidx * HEADS + hd] / denom[d * HEADS + hd];
    const float4 hv = *(const float4*)(h + (size_t)s * OUT_CH + lane * 4);
    float* op = out + (size_t)d * OUT_CH + lane * 4;
    atomicAdd(op + 0, hv.x * coef);
    atomicAdd(op + 1, hv.y * coef);
    atomicAdd(op + 2, hv.z * coef);
    atomicAdd(op + 3, hv.w * coef);
}

// ---------------------------------------------------------------------------
extern "C" void kernel_launch(void* const* d_in, const int* in_sizes, int n_in,
                              void* d_out, int out_size, void* d_ws, size_t ws_size,
                              hipStream_t stream) {
    const float* x        = (const float*)d_in[0];
    const int*   ei       = (const int*)d_in[1];
    const float* W        = (const float*)d_in[2];
    const float* att_src  = (const float*)d_in[3];
    const float* att_dst  = (const float*)d_in[4];
    const float* bias     = (const float*)d_in[5];
    float* out = (float*)d_out;

    const int n  = in_sizes[0] / IN_CH;     // 50000
    const int e  = in_sizes[1] / 2;         // 1600000
    const int ep = e + n;                   // with self loops

    // workspace partition
    char* p = (char*)d_ws;
    float*    h      = (float*)p;                 p += (size_t)n * OUT_CH * sizeof(float);
    float*    a_src  = (float*)p;                 p += (size_t)n * HEADS * sizeof(float);
    float*    a_dst  = (float*)p;                 p += (size_t)n * HEADS * sizeof(float);
    unsigned* amax   = (unsigned*)p;              p += (size_t)n * HEADS * sizeof(unsigned);
    float*    denom  = (float*)p;                 p += (size_t)n * HEADS * sizeof(float);
    float*    exbuf  = (float*)p;                 /* ep * HEADS floats */

    gat_init<<<(n * OUT_CH + 255) / 256, 256, 0, stream>>>(out, bias, amax, denom, n);
    gat_gemm<<<(n + 127) / 128, 256, 0, stream>>>(x, W, h, n);
    gat_logits<<<(n * HEADS + 255) / 256, 256, 0, stream>>>(h, att_src, att_dst,
                                                            a_src, a_dst, n);
    gat_pass1<<<(ep + 255) / 256, 256, 0, stream>>>(ei, a_src, a_dst, amax, e, n);
    gat_pass2<<<(ep + 255) / 256, 256, 0, stream>>>(ei, a_src, a_dst, amax,
                                                    exbuf, denom, e, n);
    long long threads3 = (long long)ep * 32;
    gat_pass3<<<(int)((threads3 + 255) / 256), 256, 0, stream>>>(ei, h, exbuf, denom,
                                                                 out, e, n);
}